// DTVNet_15195594293586
// MI455X (gfx1250) — compile-verified
//
#include <hip/hip_runtime.h>
#include <stdint.h>

// DTVNet cascades on (B=2, D=256, H=256, W=128) fp32.
// Bandwidth-bound: fused dual+primal kernels per cascade, p/q/s state in d_ws,
// CDNA5 async global->LDS staging for the forward-difference stencil.

#define LAMB 0.01f
#define B_ 2
#define D_ 256
#define H_ 256
#define W_ 128
#define HT 16            // H-rows per dual-kernel tile
#define NELEM ((size_t)B_ * D_ * H_ * W_)

__device__ __forceinline__ float zfun(float t, float s) {
    // z = t - LAMB*(t - sino)
    return fmaf(-LAMB, t - s, t);
}
__device__ __forceinline__ float clipf(float x, float s) {
    return fminf(fmaxf(x, -s), s);
}

// CDNA5 async copy: 16 bytes global -> LDS per lane, tracked by ASYNCcnt.
__device__ __forceinline__ void async_b128(const float* gbase, uint32_t gbyte,
                                           uint32_t ldsbyte) {
    asm volatile("global_load_async_to_lds_b128 %0, %1, %2"
                 :: "v"(ldsbyte), "v"(gbyte), "s"(gbase)
                 : "memory");
}
__device__ __forceinline__ void wait_async0() {
    asm volatile("s_wait_asynccnt 0" ::: "memory");
}

// ---------------------------------------------------------------------------
// Dual update: per element
//   z   = t - LAMB*(t - sino)
//   pn  = clip(p - (z[d+1]-z), sg0); p = pn + nt*(pn - p)   (axis D)
//   qn  = clip(q - (z[h+1]-z), sg1); q = qn + nt*(qn - q)   (axis H)
//   sn  = clip(s - (z[w+1]-z), sg2); s = sn + nt*(sn - s)   (axis W)
// Forward diff is 0 at the last index of each axis.
// t,sino tiles staged in LDS via async-to-LDS; d+1 plane staged too.
// ---------------------------------------------------------------------------
__global__ __launch_bounds__(256)
void dtv_dual(const float* __restrict__ tIn, const float* __restrict__ sino,
              float* __restrict__ P, float* __restrict__ Q, float* __restrict__ S,
              const float* __restrict__ sigma, const float* __restrict__ nt,
              int c)
{
    __shared__ __align__(16) float tT[(HT + 1) * W_];  // plane d, rows h0..h0+HT
    __shared__ __align__(16) float sT[(HT + 1) * W_];
    __shared__ __align__(16) float tD[HT * W_];        // plane d+1, rows h0..h0+HT-1
    __shared__ __align__(16) float sD[HT * W_];

    const int tid = threadIdx.x;
    const int h0  = blockIdx.x * HT;
    const int d   = blockIdx.y;
    const int b   = blockIdx.z;

    const uint32_t rowD  = (uint32_t)((b * D_ + d) * H_);
    const int dd1        = (d + 1 < D_) ? d + 1 : d;          // clamp (masked later)
    const uint32_t rowD1 = (uint32_t)((b * D_ + dd1) * H_);

    const uint32_t tTb = (uint32_t)(uintptr_t)&tT[0];
    const uint32_t sTb = (uint32_t)(uintptr_t)&sT[0];
    const uint32_t tDb = (uint32_t)(uintptr_t)&tD[0];
    const uint32_t sDb = (uint32_t)(uintptr_t)&sD[0];

    // Stage plane d: (HT+1) rows x 128 floats, 32 b128-chunks per row.
    for (int chunk = tid; chunk < (HT + 1) * (W_ / 4); chunk += 256) {
        const int row = chunk >> 5;
        const int cw  = chunk & 31;
        int hh = h0 + row; if (hh > H_ - 1) hh = H_ - 1;      // clamp (masked later)
        const uint32_t gb = ((rowD + (uint32_t)hh) * W_ + (uint32_t)cw * 4u) * 4u;
        const uint32_t lb = (uint32_t)(row * W_ + cw * 4) * 4u;
        async_b128(tIn,  gb, tTb + lb);
        async_b128(sino, gb, sTb + lb);
    }
    // Stage plane d+1: HT rows x 128 floats.
    for (int chunk = tid; chunk < HT * (W_ / 4); chunk += 256) {
        const int row = chunk >> 5;
        const int cw  = chunk & 31;
        int hh = h0 + row; if (hh > H_ - 1) hh = H_ - 1;
        const uint32_t gb = ((rowD1 + (uint32_t)hh) * W_ + (uint32_t)cw * 4u) * 4u;
        const uint32_t lb = (uint32_t)(row * W_ + cw * 4) * 4u;
        async_b128(tIn,  gb, tDb + lb);
        async_b128(sino, gb, sDb + lb);
    }
    wait_async0();
    __syncthreads();

    const float sg0 = sigma[0], sg1 = sigma[1], sg2 = sigma[2];
    const float ntc = nt[c];
    const bool  first = (c == 0);

    const int w  = tid & (W_ - 1);       // 0..127
    const int rh = tid >> 7;             // 0..1 -> row half
    const int wp = (w < W_ - 1) ? w + 1 : w;   // clamped (masked)

    const int r0 = rh * (HT / 2);
    for (int r = r0; r < r0 + HT / 2; ++r) {
        const int h = h0 + r;

        const float zc  = zfun(tT[r * W_ + w],        sT[r * W_ + w]);
        const float zd1 = zfun(tD[r * W_ + w],        sD[r * W_ + w]);
        const float zh1 = zfun(tT[(r + 1) * W_ + w],  sT[(r + 1) * W_ + w]);
        const float zw1 = zfun(tT[r * W_ + wp],       sT[r * W_ + wp]);

        const float diff1 = (d < D_ - 1) ? (zd1 - zc) : 0.0f;
        const float diff2 = (h < H_ - 1) ? (zh1 - zc) : 0.0f;
        const float diff3 = (w < W_ - 1) ? (zw1 - zc) : 0.0f;

        const size_t g = ((size_t)(b * D_ + d) * H_ + h) * W_ + w;

        const float po = first ? 0.0f : P[g];
        const float qo = first ? 0.0f : Q[g];
        const float so = first ? 0.0f : S[g];

        const float pn = clipf(po - diff1, sg0);
        const float qn = clipf(qo - diff2, sg1);
        const float sn = clipf(so - diff3, sg2);

        P[g] = pn + ntc * (pn - po);
        Q[g] = qn + ntc * (qn - qo);
        S[g] = sn + ntc * (sn - so);
    }
}

// ---------------------------------------------------------------------------
// Primal update (adjoint stencils + clipped z), float4 along W:
//   t_new = D^T p (axis D) + D^T q (axis H) + D^T s (axis W) + clip(z, sg3)
//   D^T x [j] = (j>0 ? x[j-1] : 0) - (j<N-1 ? x[j] : 0)
// ---------------------------------------------------------------------------
__global__ __launch_bounds__(256)
void dtv_primal(const float* __restrict__ tIn, const float* __restrict__ sino,
                const float* __restrict__ P, const float* __restrict__ Q,
                const float* __restrict__ S, const float* __restrict__ sigma,
                float* __restrict__ tOut)
{
    const int v = blockIdx.x * blockDim.x + threadIdx.x;   // 4-elem group id
    const int w4     = v & (W_ / 4 - 1);                   // 0..31
    const int rowIdx = v >> 5;                             // (b*D + d)*H + h
    const int h  = rowIdx & (H_ - 1);
    const int db = rowIdx >> 8;
    const int d  = db & (D_ - 1);
    const size_t g = (size_t)v * 4;

    const float4 t4 = *(const float4*)(tIn  + g);
    const float4 n4 = *(const float4*)(sino + g);
    const float4 p4 = *(const float4*)(P + g);
    const float4 q4 = *(const float4*)(Q + g);
    const float4 s4 = *(const float4*)(S + g);

    float4 pm = make_float4(0.f, 0.f, 0.f, 0.f);
    float4 qm = make_float4(0.f, 0.f, 0.f, 0.f);
    if (d > 0) pm = *(const float4*)(P + g - (size_t)H_ * W_);
    if (h > 0) qm = *(const float4*)(Q + g - W_);
    const float sprev = (w4 > 0) ? S[g - 1] : 0.0f;

    const float sg3  = sigma[3];
    const bool  dTop = (d == D_ - 1);
    const bool  hTop = (h == H_ - 1);

    const float pc[4]  = {p4.x, p4.y, p4.z, p4.w};
    const float pmv[4] = {pm.x, pm.y, pm.z, pm.w};
    const float qc[4]  = {q4.x, q4.y, q4.z, q4.w};
    const float qmv[4] = {qm.x, qm.y, qm.z, qm.w};
    const float sc[4]  = {s4.x, s4.y, s4.z, s4.w};
    const float tt[4]  = {t4.x, t4.y, t4.z, t4.w};
    const float nn[4]  = {n4.x, n4.y, n4.z, n4.w};

    float res[4];
#pragma unroll
    for (int j = 0; j < 4; ++j) {
        const int w = w4 * 4 + j;
        const float dp  = pmv[j] - (dTop ? 0.0f : pc[j]);
        const float dq  = qmv[j] - (hTop ? 0.0f : qc[j]);
        const float smj = (j == 0) ? sprev : sc[j - 1];
        const float dsv = smj - ((w < W_ - 1) ? sc[j] : 0.0f);
        const float znew = clipf(zfun(tt[j], nn[j]), sg3);
        res[j] = dp + dq + dsv + znew;
    }
    *(float4*)(tOut + g) = make_float4(res[0], res[1], res[2], res[3]);
}

extern "C" void kernel_launch(void* const* d_in, const int* in_sizes, int n_in,
                              void* d_out, int out_size, void* d_ws, size_t ws_size,
                              hipStream_t stream) {
    (void)in_sizes; (void)n_in; (void)out_size; (void)ws_size;
    const float* image = (const float*)d_in[0];
    const float* sino  = (const float*)d_in[1];
    const float* sigma = (const float*)d_in[2];
    const float* nt    = (const float*)d_in[3];
    float* out = (float*)d_out;

    // Workspace: p, q, s state (3 x 64 MB).
    float* P = (float*)d_ws;
    float* Q = P + NELEM;
    float* S = Q + NELEM;

    // outs[0] = image (stream-ordered d2d copy; graph-capture safe)
    hipMemcpyAsync(out, image, NELEM * sizeof(float),
                   hipMemcpyDeviceToDevice, stream);

    const dim3 dualGrid(H_ / HT, D_, B_);
    const dim3 dualBlock(256);
    const dim3 primGrid((unsigned)((NELEM / 4 + 255) / 256));
    const dim3 primBlock(256);

    for (int c = 0; c < 3; ++c) {
        const float* tCur = out + (size_t)c * NELEM;        // outs[c]
        float* tNext      = out + (size_t)(c + 1) * NELEM;  // outs[c+1]
        dtv_dual<<<dualGrid, dualBlock, 0, stream>>>(tCur, sino, P, Q, S,
                                                     sigma, nt, c);
        dtv_primal<<<primGrid, primBlock, 0, stream>>>(tCur, sino, P, Q, S,
                                                       sigma, tNext);
    }
}